// GATLayer_7155415515187
// MI455X (gfx1250) — compile-verified
//
#include <hip/hip_runtime.h>
#include <math.h>

typedef __attribute__((ext_vector_type(2))) float v2f;
typedef __attribute__((ext_vector_type(8))) float v8f;

#define N_DRUG 20000
#define N_TAIL 20000
#define N_RELC 64
#define NEDGE  1000000
#define DD     64
#define EPSV   1e-5f
#define NSLOPE 0.01f

// ---- workspace layout (float element offsets) ----
#define OFF_V3     0
#define OFF_V4     64
#define OFF_W1S    128
#define OFF_B1SUM  192
#define OFF_MEAN   256
#define OFF_SCALE  320
#define OFF_CSUM   384
#define OFF_CSQ    448
#define OFF_AHEAD  512
#define OFF_ATAIL  (OFF_AHEAD + N_DRUG)
#define OFF_SEGMAX (OFF_ATAIL + N_TAIL)            // stored as ordered-encoded uint
#define OFF_DENOM  (OFF_SEGMAX + N_DRUG)
#define OFF_SHR    (OFF_DENOM + N_DRUG)            // 20000*64
#define OFF_ATT    (OFF_SHR + N_DRUG * N_RELC)     // 1,000,000
#define OFF_NUMER  (OFF_ATT + NEDGE)               // 20000*64
#define OFF_Y      (OFF_NUMER + N_DRUG * DD)       // 20000*64

// monotonic float<->uint order encoding for atomicMax on signed floats
__device__ __forceinline__ unsigned enc_f32(float f) {
  unsigned u = __float_as_uint(f);
  return (u & 0x80000000u) ? ~u : (u | 0x80000000u);
}
__device__ __forceinline__ float dec_f32(unsigned u) {
  return (u & 0x80000000u) ? __uint_as_float(u & 0x7FFFFFFFu) : __uint_as_float(~u);
}

// K0: v3 = W3 @ W5[:64], v4 = W4 @ W5[64:], w1s = rowsum(W1), b1sum
__global__ void k0_precompute(const float* __restrict__ W1, const float* __restrict__ b1,
                              const float* __restrict__ W3, const float* __restrict__ W4,
                              const float* __restrict__ W5, float* __restrict__ ws) {
  int k = threadIdx.x; // 64 threads
  float v3 = 0.f, v4 = 0.f, w1s = 0.f;
  for (int j = 0; j < DD; ++j) {
    v3  += W3[k * DD + j] * W5[j];
    v4  += W4[k * DD + j] * W5[DD + j];
    w1s += W1[k * DD + j];
  }
  ws[OFF_V3 + k] = v3;
  ws[OFF_V4 + k] = v4;
  ws[OFF_W1S + k] = w1s;
  if (k == 0) {
    float s = 0.f;
    for (int j = 0; j < DD; ++j) s += b1[j];
    ws[OFF_B1SUM] = s;
  }
}

// K_init: zero accumulators, seg_max = encode(-inf)
__global__ void k_init(float* __restrict__ ws) {
  int tid = blockIdx.x * blockDim.x + threadIdx.x;
  int stride = gridDim.x * blockDim.x;
  for (int idx = tid; idx < N_DRUG * DD; idx += stride) {
    ws[OFF_NUMER + idx] = 0.f;
    if (idx < N_DRUG) {
      ws[OFF_DENOM + idx] = 0.f;
      ((unsigned*)ws)[OFF_SEGMAX + idx] = 0x007FFFFFu; // enc_f32(-inf)
    }
    if (idx < 64) { ws[OFF_CSUM + idx] = 0.f; ws[OFF_CSQ + idx] = 0.f; }
  }
}

// K1: a_head[i] = drug_emb[i]·v3 ; a_tail[i] = tail_emb[i]·v4
__global__ void k1_node_dots(const float* __restrict__ drug_emb,
                             const float* __restrict__ tail_emb,
                             float* __restrict__ ws) {
  int i = blockIdx.x * blockDim.x + threadIdx.x;
  if (i < N_DRUG) {
    float s = 0.f;
    for (int k = 0; k < DD; ++k) s += drug_emb[i * DD + k] * ws[OFF_V3 + k];
    ws[OFF_AHEAD + i] = s;
  }
  if (i < N_TAIL) {
    float s = 0.f;
    for (int k = 0; k < DD; ++k) s += tail_emb[i * DD + k] * ws[OFF_V4 + k];
    ws[OFF_ATAIL + i] = s;
  }
}

// K2: s_hr = (drug_emb ⊙ w1s) @ rel_emb^T  via V_WMMA_F32_16X16X4_F32
// one wave per 16x16 tile; M=20000 (1250 tiles), N=64 (4 tiles), K=64 (16 steps)
__global__ void k2_shr_wmma(const float* __restrict__ drug_emb,
                            const float* __restrict__ rel_emb,
                            float* __restrict__ ws) {
  const float* w1s = ws + OFF_W1S;
  float* shr = ws + OFF_SHR;
  int lane = threadIdx.x;
  int half = lane >> 4;         // 0: K pair {0,1}, rows M=0..7 in C ; 1: K pair {2,3}, rows M=8..15
  int mrow = lane & 15;
  int row0 = blockIdx.x * 16;
  int ncol = blockIdx.y * 16 + mrow;
  v8f c = {};
  for (int k0 = 0; k0 < DD; k0 += 4) {
    int ka = k0 + 2 * half;
    v2f a, b;
    a.x = drug_emb[(row0 + mrow) * DD + ka]     * w1s[ka];
    a.y = drug_emb[(row0 + mrow) * DD + ka + 1] * w1s[ka + 1];
    b.x = rel_emb[ncol * DD + ka];       // B[k][n] = rel_emb[n][k]
    b.y = rel_emb[ncol * DD + ka + 1];
    c = __builtin_amdgcn_wmma_f32_16x16x4_f32(false, a, false, b, (short)0, c, false, false);
  }
  for (int i = 0; i < 8; ++i) {
    int row = row0 + half * 8 + i;  // C layout: vgpr i = M=i (lanes 0-15) / M=8+i (lanes 16-31)
    shr[row * N_RELC + ncol] = c[i];
  }
}

// K4: att[e] = leaky_relu(a_head[h] + a_tail[t]); atomic segment max per head
__global__ void k4_edge_att(const int* __restrict__ DKG, float* __restrict__ ws) {
  int e = blockIdx.x * blockDim.x + threadIdx.x;
  if (e >= NEDGE) return;
  int h = DKG[e * 3 + 0];
  int t = DKG[e * 3 + 1];
  float x = ws[OFF_AHEAD + h] + ws[OFF_ATAIL + t];
  float att = x > 0.f ? x : NSLOPE * x;
  ws[OFF_ATT + e] = att;
  atomicMax((unsigned*)ws + OFF_SEGMAX + h, enc_f32(att));
}

// K5: one wave per edge: e=exp(att-segmax); denom[h]+=e; numer[h][:]+=e*score*t[:]
__global__ void k5_edge_accum(const int* __restrict__ DKG,
                              const float* __restrict__ tail_emb,
                              float* __restrict__ ws) {
  int wave = (blockIdx.x * blockDim.x + threadIdx.x) >> 5;
  int lane = threadIdx.x & 31;
  if (wave >= NEDGE) return;
  int h = DKG[wave * 3 + 0];
  int t = DKG[wave * 3 + 1];
  int r = DKG[wave * 3 + 2];
  float smax = dec_f32(((const unsigned*)ws)[OFF_SEGMAX + h]);
  float ev = expf(ws[OFF_ATT + wave] - smax);
  float score = ws[OFF_SHR + h * N_RELC + r] + ws[OFF_B1SUM];
  float w = ev * score;
  if (lane == 0) atomicAdd(ws + OFF_DENOM + h, ev);
  int d0 = lane * 2;
  atomicAdd(ws + OFF_NUMER + h * DD + d0,     w * tail_emb[t * DD + d0]);
  atomicAdd(ws + OFF_NUMER + h * DD + d0 + 1, w * tail_emb[t * DD + d0 + 1]);
}

// K7: y = [drug_emb | numer/denom] @ W2 + b2 via WMMA f32 16x16x4 (K=128),
// fused column sum / sumsq atomics for the batch-norm stats
__global__ void k7_y_wmma(const float* __restrict__ drug_emb,
                          const float* __restrict__ W2,
                          const float* __restrict__ b2,
                          float* __restrict__ ws) {
  int lane = threadIdx.x;
  int half = lane >> 4;
  int mrow = lane & 15;
  int row0 = blockIdx.x * 16;
  int ncol = blockIdx.y * 16 + mrow;
  const float* numer = ws + OFF_NUMER;
  const float* denom = ws + OFF_DENOM;
  v8f c = {};
  for (int k0 = 0; k0 < 2 * DD; k0 += 4) {
    int ka = k0 + 2 * half;   // ka,ka+1 never straddle the 64 boundary (k0 % 4 == 0)
    int row = row0 + mrow;
    v2f a, b;
    if (ka < DD) {
      a.x = drug_emb[row * DD + ka];
      a.y = drug_emb[row * DD + ka + 1];
    } else {
      float dn = denom[row];
      float inv = dn > 0.f ? 1.f / dn : 0.f;   // heads with no edges -> neigh = 0
      a.x = numer[row * DD + (ka - DD)] * inv;
      a.y = numer[row * DD + (ka - DD) + 1] * inv;
    }
    b.x = W2[ka * DD + ncol];
    b.y = W2[(ka + 1) * DD + ncol];
    c = __builtin_amdgcn_wmma_f32_16x16x4_f32(false, a, false, b, (short)0, c, false, false);
  }
  float bb = b2[ncol];
  float s = 0.f, sq = 0.f;
  for (int i = 0; i < 8; ++i) {
    float yv = c[i] + bb;
    ws[OFF_Y + (row0 + half * 8 + i) * DD + ncol] = yv;
    s += yv;
    sq += yv * yv;
  }
  s  += __shfl_down(s, 16, 32);
  sq += __shfl_down(sq, 16, 32);
  if (half == 0) {
    atomicAdd(ws + OFF_CSUM + ncol, s);
    atomicAdd(ws + OFF_CSQ + ncol, sq);
  }
}

// K8: per-column mean and gamma*rsqrt(var+eps)
__global__ void k8_stats(const float* __restrict__ gamma, float* __restrict__ ws) {
  int n = threadIdx.x; // 64
  float mean = ws[OFF_CSUM + n] / (float)N_DRUG;
  float var  = ws[OFF_CSQ + n] / (float)N_DRUG - mean * mean;
  ws[OFF_MEAN + n]  = mean;
  ws[OFF_SCALE + n] = gamma[n] * rsqrtf(var + EPSV);
}

// K9: out = scale*(y-mean)+beta ; second output = X passthrough
__global__ void k9_out(const float* __restrict__ X, const float* __restrict__ beta,
                       const float* __restrict__ ws, float* __restrict__ out) {
  int idx = blockIdx.x * blockDim.x + threadIdx.x;
  const int total = N_DRUG * DD;
  if (idx >= total) return;
  int n = idx & (DD - 1);
  out[idx] = ws[OFF_SCALE + n] * (ws[OFF_Y + idx] - ws[OFF_MEAN + n]) + beta[n];
  out[total + idx] = X[idx];
}

extern "C" void kernel_launch(void* const* d_in, const int* in_sizes, int n_in,
                              void* d_out, int out_size, void* d_ws, size_t ws_size,
                              hipStream_t stream) {
  const float* X        = (const float*)d_in[0];
  const int*   DKG      = (const int*)d_in[1];
  const float* drug_emb = (const float*)d_in[2];
  const float* rel_emb  = (const float*)d_in[3];
  const float* tail_emb = (const float*)d_in[4];
  const float* W1       = (const float*)d_in[5];
  const float* b1       = (const float*)d_in[6];
  const float* W2       = (const float*)d_in[7];
  const float* b2       = (const float*)d_in[8];
  const float* gamma    = (const float*)d_in[9];
  const float* beta     = (const float*)d_in[10];
  const float* W3       = (const float*)d_in[11];
  const float* W4       = (const float*)d_in[12];
  const float* W5       = (const float*)d_in[13];
  float* ws  = (float*)d_ws;
  float* out = (float*)d_out;
  (void)in_sizes; (void)n_in; (void)out_size; (void)ws_size;

  hipLaunchKernelGGL(k0_precompute, dim3(1), dim3(64), 0, stream, W1, b1, W3, W4, W5, ws);
  hipLaunchKernelGGL(k_init, dim3(1250), dim3(256), 0, stream, ws);
  hipLaunchKernelGGL(k1_node_dots, dim3((N_DRUG + 255) / 256), dim3(256), 0, stream,
                     drug_emb, tail_emb, ws);
  hipLaunchKernelGGL(k2_shr_wmma, dim3(N_DRUG / 16, N_RELC / 16), dim3(32), 0, stream,
                     drug_emb, rel_emb, ws);
  hipLaunchKernelGGL(k4_edge_att, dim3((NEDGE + 255) / 256), dim3(256), 0, stream, DKG, ws);
  hipLaunchKernelGGL(k5_edge_accum, dim3(NEDGE / 8), dim3(256), 0, stream, DKG, tail_emb, ws);
  hipLaunchKernelGGL(k7_y_wmma, dim3(N_DRUG / 16, DD / 16), dim3(32), 0, stream,
                     drug_emb, W2, b2, ws);
  hipLaunchKernelGGL(k8_stats, dim3(1), dim3(64), 0, stream, gamma, ws);
  hipLaunchKernelGGL(k9_out, dim3((N_DRUG * DD + 255) / 256), dim3(256), 0, stream,
                     X, beta, ws, out);
}